// FbpLayer_29884382446441
// MI455X (gfx1250) — compile-verified
//
#include <hip/hip_runtime.h>

typedef __attribute__((ext_vector_type(2))) float v2f;
typedef __attribute__((ext_vector_type(8))) float v8f;

#define N_ANG      360
#define N_DET      357
#define N_ROWS     1440            // B * N_ANG
#define FILT_LEN   713
#define PAD_L      356             // (FILT_LEN-1)/2
#define XPAD_STRIDE 1088           // >= 352 + 727 + 1, multiple of 16
#define FPAD_OFF   16
#define FPAD_LEN   752             // >= 727 + 2 + 16 + 1, padded
#define N_DTILE    23              // ceil(357/16)
#define N_RGRP     90              // 1440/16
#define N_PIXELS   65536
#define NNZ_TOT    8388608
#define N_COLS     (N_ANG * N_DET) // 128520

// ---------------- prep: r = sin_fan * cos, written zero-padded per row -------
__global__ void fbp_prep(const float* __restrict__ sin_fan,
                         const float* __restrict__ cosw,
                         float* __restrict__ xpad) {
  const int row = blockIdx.x;                 // 0..1439  (= b*360 + ang)
  const int ang = row % N_ANG;
  const float* s = sin_fan + (size_t)row * N_DET;
  const float* c = cosw + (size_t)ang * N_DET;
  float* xr = xpad + (size_t)row * XPAD_STRIDE;
  for (int j = threadIdx.x; j < XPAD_STRIDE; j += blockDim.x) {
    int d = j - PAD_L;
    float v = 0.0f;
    if ((unsigned)d < (unsigned)N_DET) v = s[d] * c[d];
    xr[j] = v;
  }
}

// ---------------- conv as Toeplitz GEMM on V_WMMA_F32_16X16X4_F32 ------------
// out[m, d0+n] = sum_{k=0..727} xpad[m, d0+k] * filt[k-n]
__global__ void __launch_bounds__(32)
fbp_conv_wmma(const float* __restrict__ xpad,
              const float* __restrict__ filt,
              float* __restrict__ Xmat /* [N_COLS][4] */) {
  __shared__ float fpad[FPAD_LEN];            // fpad[j] = filt[j-16], zero elsewhere
  const int lane = threadIdx.x;
  for (int j = lane; j < FPAD_LEN; j += 32) {
    int fi = j - FPAD_OFF;
    fpad[j] = ((unsigned)fi < (unsigned)FILT_LEN) ? filt[fi] : 0.0f;
  }
  __syncthreads();

  const int tile    = blockIdx.x;
  const int rg      = tile / N_DTILE;
  const int dt      = tile % N_DTILE;
  const int rowbase = rg * 16;
  const int d0      = dt * 16;

  const int m   = lane & 15;                  // A-matrix row for this lane
  const int kkb = (lane >> 4) << 1;           // 0 (lanes 0-15) or 2 (lanes 16-31)
  const int n   = lane & 15;                  // B/D column for this lane

  const float* arow = xpad + (size_t)(rowbase + m) * XPAD_STRIDE + d0 + kkb;
  const float* frow = fpad + (FPAD_OFF + kkb - n);   // min index = 1, max = 743

  v8f acc = {};
  #pragma unroll 2
  for (int k = 0; k < 728; k += 4) {
    v2f a = *(const v2f*)(arow + k);          // A: K = kkb, kkb+1 (8B aligned)
    v2f b;
    b.x = frow[k];                            // B: K = kkb   row, col n
    b.y = frow[k + 1];                        // B: K = kkb+1 row, col n
    acc = __builtin_amdgcn_wmma_f32_16x16x4_f32(
        /*neg_a=*/false, a, /*neg_b=*/false, b,
        /*c_mod=*/(short)0, acc, /*reuse_a=*/false, /*reuse_b=*/false);
  }

  // D layout: VGPR v -> M = v + 8*(lane>=16), N = lane&15
  const int mout = (lane >> 4) << 3;
  const int d = d0 + n;
  if (d < N_DET) {
    #pragma unroll
    for (int v = 0; v < 8; ++v) {
      int row   = rowbase + mout + v;
      int b_idx = row / N_ANG;
      int ang   = row % N_ANG;
      Xmat[((size_t)ang * N_DET + d) * 4 + b_idx] = acc[v];
    }
  }
}

// ---------------- zero the fp32 accumulator ----------------------------------
__global__ void fbp_zero(float* __restrict__ acc, int n) {
  int i = blockIdx.x * blockDim.x + threadIdx.x;
  if (i < n) acc[i] = 0.0f;
}

// ---------------- sparse gather + segment-sum via global fp32 atomics --------
__global__ void fbp_scatter(const float* __restrict__ vals,
                            const int* __restrict__ rows,
                            const int* __restrict__ cols,
                            const float4* __restrict__ Xmat,
                            float* __restrict__ acc) {
  int t = blockIdx.x * blockDim.x + threadIdx.x;   // 0 .. NNZ/4-1
  int i0 = t << 2;
  float4 v = *(const float4*)(vals + i0);
  int4   r = *(const int4*)(rows + i0);
  int4   c = *(const int4*)(cols + i0);

  float4 x;
  x = Xmat[c.x];
  atomicAdd(acc + 4 * r.x + 0, v.x * x.x);
  atomicAdd(acc + 4 * r.x + 1, v.x * x.y);
  atomicAdd(acc + 4 * r.x + 2, v.x * x.z);
  atomicAdd(acc + 4 * r.x + 3, v.x * x.w);
  x = Xmat[c.y];
  atomicAdd(acc + 4 * r.y + 0, v.y * x.x);
  atomicAdd(acc + 4 * r.y + 1, v.y * x.y);
  atomicAdd(acc + 4 * r.y + 2, v.y * x.z);
  atomicAdd(acc + 4 * r.y + 3, v.y * x.w);
  x = Xmat[c.z];
  atomicAdd(acc + 4 * r.z + 0, v.z * x.x);
  atomicAdd(acc + 4 * r.z + 1, v.z * x.y);
  atomicAdd(acc + 4 * r.z + 2, v.z * x.z);
  atomicAdd(acc + 4 * r.z + 3, v.z * x.w);
  x = Xmat[c.w];
  atomicAdd(acc + 4 * r.w + 0, v.w * x.x);
  atomicAdd(acc + 4 * r.w + 1, v.w * x.y);
  atomicAdd(acc + 4 * r.w + 2, v.w * x.z);
  atomicAdd(acc + 4 * r.w + 3, v.w * x.w);
}

// ---------------- transpose (pix, B) -> (B, pix) with clip -------------------
__device__ __forceinline__ float clip01(float v) {
  return fminf(fmaxf(v, 0.0f), 1.0f);
}

__global__ void fbp_final(const float4* __restrict__ acc, float* __restrict__ out) {
  int p = blockIdx.x * blockDim.x + threadIdx.x;
  if (p < N_PIXELS) {
    float4 a = acc[p];
    out[0 * N_PIXELS + p] = clip01(a.x);
    out[1 * N_PIXELS + p] = clip01(a.y);
    out[2 * N_PIXELS + p] = clip01(a.z);
    out[3 * N_PIXELS + p] = clip01(a.w);
  }
}

extern "C" void kernel_launch(void* const* d_in, const int* in_sizes, int n_in,
                              void* d_out, int out_size, void* d_ws, size_t ws_size,
                              hipStream_t stream) {
  const float* sin_fan = (const float*)d_in[0];
  const float* cosw    = (const float*)d_in[1];
  const float* filt    = (const float*)d_in[2];
  const float* A_vals  = (const float*)d_in[3];
  const int*   A_rows  = (const int*)d_in[4];
  const int*   A_cols  = (const int*)d_in[5];
  float* out = (float*)d_out;

  // workspace layout (all 16B aligned):
  //   xpad : N_ROWS * XPAD_STRIDE floats            (6,266,880 B)
  //   Xmat : N_COLS * 4 floats (transposed sino)    (2,056,320 B)
  //   acc  : N_PIXELS * 4 floats                    (1,048,576 B)
  char* ws = (char*)d_ws;
  float* xpad = (float*)ws;
  float* Xmat = (float*)(ws + (size_t)N_ROWS * XPAD_STRIDE * sizeof(float));
  float* acc  = (float*)(ws + (size_t)N_ROWS * XPAD_STRIDE * sizeof(float)
                            + (size_t)N_COLS * 4 * sizeof(float));

  fbp_prep<<<N_ROWS, 256, 0, stream>>>(sin_fan, cosw, xpad);
  fbp_conv_wmma<<<N_RGRP * N_DTILE, 32, 0, stream>>>(xpad, filt, Xmat);
  fbp_zero<<<(N_PIXELS * 4 + 255) / 256, 256, 0, stream>>>(acc, N_PIXELS * 4);
  fbp_scatter<<<(NNZ_TOT / 4) / 256, 256, 0, stream>>>(A_vals, A_rows, A_cols,
                                                       (const float4*)Xmat, acc);
  fbp_final<<<(N_PIXELS + 255) / 256, 256, 0, stream>>>((const float4*)acc, out);
}